// GPSNodeClassifier_14525579395562
// MI455X (gfx1250) — compile-verified
//
#include <hip/hip_runtime.h>

#define NNODES 50000
#define NEDGES 640000
#define FDIM   128
#define CLSDIM 64
#define MTILES (NNODES / 16)   // 3125, exact

typedef float v2f __attribute__((ext_vector_type(2)));
typedef float v8f __attribute__((ext_vector_type(8)));

// ---------------- utility kernels ----------------

__global__ __launch_bounds__(256) void fill_kernel(float* __restrict__ p, float v, int n) {
    int t = blockIdx.x * 256 + threadIdx.x;
    if (t < n) p[t] = v;
}

__global__ __launch_bounds__(256) void deg_count_kernel(const int* __restrict__ col,
                                                        float* __restrict__ deg, int e) {
    int t = blockIdx.x * 256 + threadIdx.x;
    if (t < e) atomicAdd(&deg[col[t]], 1.0f);
}

__global__ __launch_bounds__(256) void rsqrt_kernel(float* __restrict__ d, int n) {
    int t = blockIdx.x * 256 + threadIdx.x;
    if (t < n) d[t] = rsqrtf(d[t]);
}

__global__ __launch_bounds__(256) void enorm_kernel(const int* __restrict__ row,
                                                    const int* __restrict__ col,
                                                    const float* __restrict__ dinv,
                                                    float* __restrict__ enorm, int e) {
    int t = blockIdx.x * 256 + threadIdx.x;
    if (t < e) enorm[t] = dinv[row[t]] * dinv[col[t]];
}

// ---------------- WMMA fp32 GEMM: Out[M x NCOLS] = act(A)[M x 128] @ W[128 x NCOLS] (+bias) ----
// One wave computes a 16-row strip across all NCOLS columns (8 or 4 accumulator tiles).
// Block = 256 threads = 8 waves => 128 rows per block.
// W staged in LDS pair-swizzled: sW[p*RS + n*2 + {0,1}] = W[2p][n], W[2p+1][n]
// RS = 2*NCOLS+32 floats => pair-row bank offset 32 => lanes 0-15 / 16-31 use disjoint banks.
// k-loop fully unrolled: no back-edge => no accumulator register rotation (v_mov) overhead.

template <int NCOLS, bool RELU_A>
__global__ __launch_bounds__(256) void wmma_gemm_f32(const float* __restrict__ A,
                                                     const float* __restrict__ W,
                                                     const float* __restrict__ bias,
                                                     float* __restrict__ Out,
                                                     int m_tiles) {
    constexpr int RS = NCOLS * 2 + 32;
    __shared__ float sW[64 * RS];

    const int tid = threadIdx.x;
    for (int idx = tid; idx < 64 * NCOLS; idx += 256) {
        int p = idx / NCOLS;
        int n = idx - p * NCOLS;
        v2f w;
        w.x = W[(2 * p + 0) * NCOLS + n];
        w.y = W[(2 * p + 1) * NCOLS + n];
        *(v2f*)&sW[p * RS + n * 2] = w;
    }
    __syncthreads();

    const int wave = tid >> 5;
    const int lane = tid & 31;
    const int mt = blockIdx.x * 8 + wave;
    if (mt >= m_tiles) return;          // wave-uniform: EXEC stays all-ones for WMMA

    const int half = lane >> 4;         // 0: K pair {k,k+1}, 1: K pair {k+2,k+3}
    const int l16  = lane & 15;
    const float* Arow = A + (size_t)(mt * 16 + l16) * FDIM + 2 * half;

    constexpr int NT = NCOLS / 16;      // column tiles (8 or 4)
    v8f acc[NT];
#pragma unroll
    for (int j = 0; j < NT; ++j) acc[j] = (v8f){0.f, 0.f, 0.f, 0.f, 0.f, 0.f, 0.f, 0.f};

    auto do_step = [&](v2f a, int k) {
        if (RELU_A) {
            a.x = fmaxf(a.x, 0.0f);
            a.y = fmaxf(a.y, 0.0f);
        }
        const int p = (k >> 1) + half;              // K pair index for this lane
        const float* base = &sW[p * RS + l16 * 2];
#pragma unroll
        for (int j = 0; j < NT; ++j) {
            v2f b = *(const v2f*)&base[j * 32];     // single aligned ds_load_b64
            acc[j] = __builtin_amdgcn_wmma_f32_16x16x4_f32(
                false, a, false, b, (short)0, acc[j], false, false);
        }
    };

    // software-pipeline A one k-step ahead; fully unrolled (straight-line body)
    v2f a_cur = *(const v2f*)&Arow[0];
#pragma unroll
    for (int k = 0; k < FDIM - 4; k += 4) {
        v2f a_next = *(const v2f*)&Arow[k + 4];
        do_step(a_cur, k);
        a_cur = a_next;
    }
    do_step(a_cur, FDIM - 4);

    // C/D layout: VGPR r, lanes 0-15 -> M=r, lanes 16-31 -> M=r+8; N = l16
#pragma unroll
    for (int j = 0; j < NT; ++j) {
        const int n = j * 16 + l16;
        const float bv = bias ? bias[n] : 0.0f;
#pragma unroll
        for (int r = 0; r < 8; ++r) {
            const int m = mt * 16 + r + 8 * half;
            Out[(size_t)m * NCOLS + n] = acc[j][r] + bv;
        }
    }
}

// ---------------- self-loop init: agg = hw*dinv^2 + bias (replaces zero-fill) ----------------

__global__ __launch_bounds__(256) void init_self_kernel(const float* __restrict__ hw,
                                                        const float* __restrict__ dinv,
                                                        const float* __restrict__ bias,
                                                        float* __restrict__ agg, int total) {
    int t = blockIdx.x * 256 + threadIdx.x;
    if (t >= total) return;
    int n = t >> 7;     // /128
    int f = t & 127;
    float di = dinv[n];
    agg[t] = hw[t] * (di * di) + bias[f];
}

// ---------------- edge scatter: agg[col] += hw[row] * enorm, one wave per edge ----------------

__global__ __launch_bounds__(256) void scatter_kernel(const float* __restrict__ hw,
                                                      const int* __restrict__ row,
                                                      const int* __restrict__ col,
                                                      const float* __restrict__ enorm,
                                                      float* __restrict__ agg, int e) {
    int t = blockIdx.x * 256 + threadIdx.x;
    int edge = t >> 5;
    if (edge >= e) return;
    int lane = t & 31;
    int r = row[edge];
    int c = col[edge];
    float en = enorm[edge];
    float4 v = *(const float4*)(hw + (size_t)r * FDIM + lane * 4);
    float* dst = agg + (size_t)c * FDIM + lane * 4;
    atomicAdd(dst + 0, v.x * en);
    atomicAdd(dst + 1, v.y * en);
    atomicAdd(dst + 2, v.z * en);
    atomicAdd(dst + 3, v.w * en);
}

// ---------------- host-side launch ----------------

extern "C" void kernel_launch(void* const* d_in, const int* in_sizes, int n_in,
                              void* d_out, int out_size, void* d_ws, size_t ws_size,
                              hipStream_t stream) {
    const float* x    = (const float*)d_in[0];
    const int*   ei   = (const int*)d_in[1];
    const float* W0   = (const float*)d_in[2];
    const float* b0   = (const float*)d_in[3];
    const float* W1   = (const float*)d_in[4];
    const float* b1   = (const float*)d_in[5];
    const float* W2   = (const float*)d_in[6];
    const float* b2   = (const float*)d_in[7];
    const float* Wl   = (const float*)d_in[8];
    const float* bl   = (const float*)d_in[9];
    (void)in_sizes; (void)n_in; (void)out_size; (void)ws_size;

    const int* row = ei;             // edge_index[0]
    const int* col = ei + NEDGES;    // edge_index[1]

    float* ws    = (float*)d_ws;
    float* dinv  = ws;                               // 50000  (pad to 50176)
    float* enorm = ws + 50176;                       // 640000
    float* buf0  = ws + 690176;                      // 50000*128 = hw
    float* buf1  = ws + 7090176;                     // 50000*128 = agg / h (pre-relu)

    const int NF = NNODES * FDIM;                    // 6,400,000

    // --- graph normalization (graph fixed across layers; compute once per call) ---
    fill_kernel<<<(NNODES + 255) / 256, 256, 0, stream>>>(dinv, 1.0f, NNODES);
    deg_count_kernel<<<(NEDGES + 255) / 256, 256, 0, stream>>>(col, dinv, NEDGES);
    rsqrt_kernel<<<(NNODES + 255) / 256, 256, 0, stream>>>(dinv, NNODES);
    enorm_kernel<<<(NEDGES + 255) / 256, 256, 0, stream>>>(row, col, dinv, enorm, NEDGES);

    const int gemm_blocks    = (MTILES + 7) / 8;             // 391
    const int scatter_blocks = (NEDGES * 32 + 255) / 256;    // 80000
    const int elem_blocks    = (NF + 255) / 256;             // 25000

    // Layer 0: input is x (no activation on A)
    wmma_gemm_f32<FDIM, false><<<gemm_blocks, 256, 0, stream>>>(x, W0, nullptr, buf0, MTILES);
    init_self_kernel<<<elem_blocks, 256, 0, stream>>>(buf0, dinv, b0, buf1, NF);
    scatter_kernel<<<scatter_blocks, 256, 0, stream>>>(buf0, row, col, enorm, buf1, NEDGES);

    // Layer 1: buf1 holds pre-relu h; relu fused into A-operand load
    wmma_gemm_f32<FDIM, true><<<gemm_blocks, 256, 0, stream>>>(buf1, W1, nullptr, buf0, MTILES);
    init_self_kernel<<<elem_blocks, 256, 0, stream>>>(buf0, dinv, b1, buf1, NF);
    scatter_kernel<<<scatter_blocks, 256, 0, stream>>>(buf0, row, col, enorm, buf1, NEDGES);

    // Layer 2
    wmma_gemm_f32<FDIM, true><<<gemm_blocks, 256, 0, stream>>>(buf1, W2, nullptr, buf0, MTILES);
    init_self_kernel<<<elem_blocks, 256, 0, stream>>>(buf0, dinv, b2, buf1, NF);
    scatter_kernel<<<scatter_blocks, 256, 0, stream>>>(buf0, row, col, enorm, buf1, NEDGES);

    // Final classifier: out = relu(h) @ W_lin + b_lin
    wmma_gemm_f32<CLSDIM, true><<<gemm_blocks, 256, 0, stream>>>(buf1, Wl, bl, (float*)d_out, MTILES);
}